// BoundaryLoss_52201032516261
// MI455X (gfx1250) — compile-verified
//
#include <hip/hip_runtime.h>
#include <math.h>

typedef __attribute__((ext_vector_type(2))) float v2f;
typedef __attribute__((ext_vector_type(4))) float v4f;
typedef __attribute__((ext_vector_type(8))) float v8f;
typedef __attribute__((ext_vector_type(4))) int   v4i;

typedef __attribute__((address_space(1))) v4i* v4i_g;   // global
typedef __attribute__((address_space(3))) v4i* v4i_l;   // LDS

#define HH 1024
#define WW 1024
#define RPB 8          // rows per block
#define TPB 256        // 8 waves of 32

// ---------------- CDNA5 async Global->LDS staging ----------------
__device__ __forceinline__ void async_copy_b128(const int* __restrict__ g, int* l) {
#if __has_builtin(__builtin_amdgcn_global_load_async_to_lds_b128)
    __builtin_amdgcn_global_load_async_to_lds_b128(
        (v4i_g)(g), (v4i_l)(l), /*imm offset*/0, /*cpol*/0);
#else
    *(int4*)l = *(const int4*)g;
#endif
}

__device__ __forceinline__ void async_wait_all() {
#if __has_builtin(__builtin_amdgcn_global_load_async_to_lds_b128)
#if __has_builtin(__builtin_amdgcn_s_wait_asynccnt)
    __builtin_amdgcn_s_wait_asynccnt(0);
#else
    asm volatile("s_wait_asynccnt 0" ::: "memory");
#endif
#endif
}

// ---------------- WMMA-based full-wave (32 lane) sum ----------------
// A(16x4): lanes0-15 {K0,K1}, lanes16-31 {K2,K3}. Set A={v,0} per lane,
// B=ones -> D[m][n] = v_m + v_{m+16}. D layout: lane<16 VGPR r = D[r][lane],
// lane>=16 VGPR r = D[8+r][lane-16]. Sum 8 regs + shfl_xor(16) = wave total.
__device__ __forceinline__ float wave_sum_wmma(float v) {
    v2f a; a[0] = v;    a[1] = 0.0f;
    v2f b; b[0] = 1.0f; b[1] = 1.0f;
    v8f c = {};
    v8f d = __builtin_amdgcn_wmma_f32_16x16x4_f32(
        /*neg_a=*/false, a, /*neg_b=*/false, b,
        /*c_mod=*/(short)0, c, /*reuse_a=*/false, /*reuse_b=*/false);
    float t = ((d[0] + d[1]) + (d[2] + d[3])) + ((d[4] + d[5]) + (d[6] + d[7]));
    t += __shfl_xor(t, 16, 32);
    return t;
}

// Fast stable BCE-with-logits term: max(x,0) - x*z + log1p(exp(-|x|)).
// Uses hardware v_exp_f32 / v_log_f32; |error| per element < ~1e-7 absolute.
__device__ __forceinline__ float bce_term(float x, float z) {
    float t = __expf(-fabsf(x));            // v_exp_f32 (exp2 scaled)
    return fmaxf(x, 0.0f) - x * z + __logf(1.0f + t);  // v_log_f32 scaled
}

// ---------------- Kernel 1: per-block partial loss ----------------
__global__ __launch_bounds__(TPB) void boundary_loss_partial(
    const float* __restrict__ pred, const int* __restrict__ gt,
    float* __restrict__ partial) {
    __shared__ int tile[(RPB + 2) * WW];   // 40 KB
    __shared__ float red[TPB / 32];

    const int tid = threadIdx.x;
    const int groupsPerImage = HH / RPB;            // 128
    const int b  = blockIdx.x / groupsPerImage;
    const int rg = blockIdx.x % groupsPerImage;
    const int row0 = rg * RPB;

    const size_t imgBase = (size_t)b * HH * WW;
    const int c0 = tid * 4;                         // 256*4 == 1024 cols

    // Stage mask rows row0-1 .. row0+RPB (clamped: clamp == SAME for min/max)
    #pragma unroll
    for (int j = 0; j < RPB + 2; ++j) {
        int sr = row0 - 1 + j;
        sr = sr < 0 ? 0 : (sr >= HH ? HH - 1 : sr);
        async_copy_b128(gt + imgBase + (size_t)sr * WW + c0, &tile[j * WW + c0]);
    }
    async_wait_all();
    __syncthreads();

    const int cl = (c0 == 0) ? 0 : (c0 - 1);
    const int cr = (c0 + 4 >= WW) ? (WW - 1) : (c0 + 4);

    float acc = 0.0f;
    #pragma unroll
    for (int i = 0; i < RPB; ++i) {
        const int* rA = &tile[(i + 0) * WW];
        const int* rB = &tile[(i + 1) * WW];
        const int* rC = &tile[(i + 2) * WW];

        // 3-row vertical OR/AND (binary max/min) for 6 columns cl,c0..c0+3,cr
        int4 tA = *(const int4*)(rA + c0);
        int4 tB = *(const int4*)(rB + c0);
        int4 tC = *(const int4*)(rC + c0);
        int lO = rA[cl] | rB[cl] | rC[cl];
        int lA = rA[cl] & rB[cl] & rC[cl];
        int rO = rA[cr] | rB[cr] | rC[cr];
        int rN = rA[cr] & rB[cr] & rC[cr];

        int orv[6]  = { lO, tA.x | tB.x | tC.x, tA.y | tB.y | tC.y,
                            tA.z | tB.z | tC.z, tA.w | tB.w | tC.w, rO };
        int andv[6] = { lA, tA.x & tB.x & tC.x, tA.y & tB.y & tC.y,
                            tA.z & tB.z & tC.z, tA.w & tB.w & tC.w, rN };

        // Streaming (non-temporal) pred load: data is read exactly once.
        const v4f* pp = (const v4f*)(pred + imgBase + (size_t)(row0 + i) * WW + c0);
        v4f px = __builtin_nontemporal_load(pp);

        #pragma unroll
        for (int p = 0; p < 4; ++p) {
            int o3 = orv[p]  | orv[p + 1]  | orv[p + 2];
            int a3 = andv[p] & andv[p + 1] & andv[p + 2];
            float z = (float)((o3 & ~a3) & 1);       // boundary iff OR=1 & AND=0
            acc += bce_term(px[p], z);
        }
    }

    // Block reduction: WMMA per wave, then LDS combine (EXEC all ones here).
    float w = wave_sum_wmma(acc);
    if ((tid & 31) == 0) red[tid >> 5] = w;
    __syncthreads();
    if (tid == 0) {
        float s = 0.0f;
        #pragma unroll
        for (int i = 0; i < TPB / 32; ++i) s += red[i];
        partial[blockIdx.x] = s;
    }
}

// ---------------- Kernel 2: deterministic final reduce ----------------
__global__ __launch_bounds__(TPB) void boundary_loss_final(
    const float* __restrict__ partial, float* __restrict__ out,
    int n, float invN) {
    __shared__ float red[TPB / 32];
    float acc = 0.0f;
    for (int i = threadIdx.x; i < n; i += TPB) acc += partial[i];  // n % 256 == 0
    float w = wave_sum_wmma(acc);
    if ((threadIdx.x & 31) == 0) red[threadIdx.x >> 5] = w;
    __syncthreads();
    if (threadIdx.x == 0) {
        float s = 0.0f;
        #pragma unroll
        for (int i = 0; i < TPB / 32; ++i) s += red[i];
        out[0] = s * invN;
    }
}

extern "C" void kernel_launch(void* const* d_in, const int* in_sizes, int n_in,
                              void* d_out, int out_size, void* d_ws, size_t ws_size,
                              hipStream_t stream) {
    const float* pred = (const float*)d_in[0];
    const int*   gt   = (const int*)d_in[1];
    float* out   = (float*)d_out;
    float* parts = (float*)d_ws;

    const long long n = (long long)in_sizes[0];          // 64 * 1024 * 1024
    const int Bimg    = (int)(n / ((long long)HH * WW)); // 64
    const int nblocks = Bimg * (HH / RPB);               // 8192

    boundary_loss_partial<<<nblocks, TPB, 0, stream>>>(pred, gt, parts);

    const float invN = 1.0f / (float)n;                  // N = 2^26, exact
    boundary_loss_final<<<1, TPB, 0, stream>>>(parts, out, nblocks, invN);
}